// EXAM_24352464569853
// MI455X (gfx1250) — compile-verified
//
#include <hip/hip_runtime.h>
#include <hip/hip_bf16.h>
#include <math.h>

#define VOCA        30000
#define EMB         128
#define REGION      7
#define RADIUS      3
#define NUM_CLASSES 20
#define MAX_LEN     512
#define ENTIRE      506          // MAX_LEN - 2*RADIUS
#define ENTIRE_PAD  512          // padded K / n dimension
#define CLASS_PAD   32           // padded class rows in X2
#define BATCH       64
#define AGG_HID     1024         // MAX_LEN*2
#define BC          (BATCH * NUM_CLASSES)  // 1280

typedef __attribute__((ext_vector_type(2))) float v2f;
typedef __attribute__((ext_vector_type(4))) float v4f;
typedef __attribute__((ext_vector_type(8))) float v8f;

static __device__ __forceinline__ v8f wmma4(v2f a, v2f b, v8f c) {
  return __builtin_amdgcn_wmma_f32_16x16x4_f32(false, a, false, b,
                                               (short)0, c, false, false);
}

// ---------------------------------------------------------------------------
// K1: P[b*506+n][e] = relu( max_r emb_word[train[b][n+r]][e]
//                                 * emb_region[train[b][n+3]][r*128+e] )
// one 128-thread block per (b, n); pure gather stage (bandwidth bound:
// ~232 MB of gathers -> this stage IS the roofline for the whole problem).
// ---------------------------------------------------------------------------
__global__ __launch_bounds__(EMB)
void k_gather_max(const int* __restrict__ train,
                  const float* __restrict__ emb_region,
                  const float* __restrict__ emb_word,
                  float* __restrict__ P) {
  int bn = blockIdx.x;
  int b  = bn / ENTIRE;
  int n  = bn % ENTIRE;
  int e  = threadIdx.x;
  const int* row = train + b * MAX_LEN;
  int tc = row[n + RADIUS];
  const float* kreg = emb_region + (size_t)tc * (REGION * EMB) + e;
  float m = -INFINITY;
  #pragma unroll
  for (int r = 0; r < REGION; ++r) {
    int   tw = row[n + r];
    float ev = emb_word[(size_t)tw * EMB + e];
    float kv = kreg[r * EMB];
    m = fmaxf(m, ev * kv);
  }
  P[(size_t)bn * EMB + e] = fmaxf(m, 0.0f);
}

// ---------------------------------------------------------------------------
// K1b: pad W_agg1 (1024 x 506) -> Wp (1024 x 512), zero tail columns.
// Branchless (clamped load + select). One-shot, 4 MB traffic.
// ---------------------------------------------------------------------------
__global__ __launch_bounds__(128)
void k_pad_w(const float* __restrict__ W_agg1, float* __restrict__ Wp) {
  int r = blockIdx.x;                  // 0..1023
  int c = threadIdx.x;                 // 0..127
  const float* src = W_agg1 + (size_t)r * ENTIRE;
  float*       dst = Wp     + (size_t)r * ENTIRE_PAD;
  #pragma unroll
  for (int j = 0; j < 4; ++j) {
    int   cc = c + j * 128;
    int   ccC = cc < ENTIRE ? cc : ENTIRE - 1;   // clamp (stay in row)
    float v  = src[ccC];
    dst[cc] = (cc < ENTIRE) ? v : 0.0f;          // branchless zero pad
  }
}

// ---------------------------------------------------------------------------
// K2: per batch b: X2P[b][c][n] = sum_e P[b*506+n][e] * W_inter[c][e]
// X2P is padded [64][32][512]: class rows 20..31 are garbage (never read),
// n columns 506..511 are zeroed in-register -> stores are UNCONDITIONAL,
// two b128 per lane (each lane's 8 acc values are 8 contiguous n's).
// A rows / B cols are clamped, never zeroed: garbage only reaches D entries
// that are zeroed at store (n>=506) or land in garbage rows (c>=20).
// grid = BATCH * 32(row tiles over n) * 2(col tiles over classes).
// ---------------------------------------------------------------------------
__global__ __launch_bounds__(32)
void k_inter_wmma(const float* __restrict__ P,
                  const float* __restrict__ W_inter,
                  float* __restrict__ X2P) {
  int bid  = blockIdx.x;
  int b    = bid >> 6;
  int rem  = bid & 63;
  int mt   = rem >> 1;                 // 0..31 row tiles over n (covers 512)
  int nt   = rem & 1;                  // 0..1 col tiles over classes
  int lane = threadIdx.x;
  int l15  = lane & 15;
  int kh   = (lane >> 4) << 1;         // K sub-offset: lanes 0-15 -> 0, 16-31 -> 2

  int nrow  = mt * 16 + l15;
  int nrowC = nrow < ENTIRE ? nrow : ENTIRE - 1;            // clamp
  int ccol  = nt * 16 + l15;                                // 0..31 (all stored)
  int ccolC = ccol < NUM_CLASSES ? ccol : NUM_CLASSES - 1;  // clamp

  const float* Arow = P + ((size_t)b * ENTIRE + nrowC) * EMB + kh;
  const float* Brow = W_inter + (size_t)ccolC * EMB + kh;

  v8f acc = {};
  #pragma unroll 4
  for (int k = 0; k < EMB; k += 4) {   // 32 WMMA steps, b64 loads
    v2f a  = *(const v2f*)(Arow + k);
    v2f bb = *(const v2f*)(Brow + k);
    acc = wmma4(a, bb, acc);
  }

  // D layout: VGPR v holds M=v (lanes 0-15) / M=v+8 (lanes 16-31), N=lane&15.
  // Lane's 8 values cover n = nbase..nbase+7 (contiguous).
  int hi    = lane >> 4;
  int nbase = mt * 16 + 8 * hi;
  float* dst = X2P + ((size_t)b * CLASS_PAD + ccol) * ENTIRE_PAD + nbase;
  v4f lo, hv;
  #pragma unroll
  for (int j = 0; j < 4; ++j) {
    lo[j] = (nbase + j     < ENTIRE) ? acc[j]     : 0.0f;   // branchless zero pad
    hv[j] = (nbase + 4 + j < ENTIRE) ? acc[4 + j] : 0.0f;
  }
  *(v4f*)(dst)     = lo;               // 16B-aligned b128 stores
  *(v4f*)(dst + 4) = hv;
}

// ---------------------------------------------------------------------------
// K3: H[bc][m] = relu( sum_n X2[bc][n] * W_agg1[m][n] + b_agg1[m] )
// bc in [0,1280), m in [0,1024), padded K = 512 (cols 506..511 are zero in
// both X2P and Wp) -> 128 clean k-steps, no tail, no masks, no guards.
// 2x2 register tiling: 2 A + 2 B b64 loads and 4 independent WMMAs per step.
// Logical row r maps to padded X2P row (r/20)*32 + r%20 (computed once).
// grid = 40(row tiles of 32 over bc) * 32(col tiles of 32 over 1024).
// ---------------------------------------------------------------------------
__global__ __launch_bounds__(32)
void k_agg1_wmma(const float* __restrict__ X2P,
                 const float* __restrict__ Wp,
                 const float* __restrict__ b_agg1,
                 float* __restrict__ H) {
  int bid  = blockIdx.x;
  int mt   = bid >> 5;                 // 0..39
  int nt   = bid & 31;                 // 0..31
  int lane = threadIdx.x;
  int l15  = lane & 15;
  int kh   = (lane >> 4) << 1;

  int r0  = mt * 32 + l15;             // logical bc rows (exact: 40*32=1280)
  int r1  = r0 + 16;
  int pr0 = (r0 / 20) * CLASS_PAD + (r0 % 20);   // padded X2P rows
  int pr1 = (r1 / 20) * CLASS_PAD + (r1 % 20);

  const float* A0 = X2P + (size_t)pr0 * ENTIRE_PAD + kh;
  const float* A1 = X2P + (size_t)pr1 * ENTIRE_PAD + kh;
  const float* B0 = Wp + (size_t)(nt * 32 + l15) * ENTIRE_PAD + kh;  // cols exact
  const float* B1 = B0 + (size_t)16 * ENTIRE_PAD;

  v8f acc00 = {}, acc01 = {}, acc10 = {}, acc11 = {};

  #pragma unroll 2
  for (int k = 0; k < ENTIRE_PAD; k += 4) {      // 128 clean steps
    v2f a0 = *(const v2f*)(A0 + k);
    v2f a1 = *(const v2f*)(A1 + k);
    v2f b0 = *(const v2f*)(B0 + k);
    v2f b1 = *(const v2f*)(B1 + k);
    acc00 = wmma4(a0, b0, acc00);
    acc01 = wmma4(a0, b1, acc01);
    acc10 = wmma4(a1, b0, acc10);
    acc11 = wmma4(a1, b1, acc11);
  }

  int   c0 = nt * 32 + l15;
  int   c1 = c0 + 16;
  float bias0 = b_agg1[c0];
  float bias1 = b_agg1[c1];
  int   hi = lane >> 4;
  #pragma unroll
  for (int v = 0; v < 8; ++v) {
    int rr0 = mt * 32 + v + 8 * hi;    // logical H rows (H stays unpadded)
    int rr1 = rr0 + 16;
    H[(size_t)rr0 * AGG_HID + c0] = fmaxf(acc00[v] + bias0, 0.0f);
    H[(size_t)rr0 * AGG_HID + c1] = fmaxf(acc01[v] + bias1, 0.0f);
    H[(size_t)rr1 * AGG_HID + c0] = fmaxf(acc10[v] + bias0, 0.0f);
    H[(size_t)rr1 * AGG_HID + c1] = fmaxf(acc11[v] + bias1, 0.0f);
  }
}

// ---------------------------------------------------------------------------
// K4: agreg[b][c] = dot(H[bc], W_agg2) + b_agg2; softmax over c; argmax.
// One block per batch row; one wave per class; wave32 shuffle reduction.
// out layout: [agreg 1280 | prob 1280 | class 64 (as float)]
// ---------------------------------------------------------------------------
__global__ __launch_bounds__(NUM_CLASSES * 32)
void k_agg2_softmax(const float* __restrict__ H,
                    const float* __restrict__ W_agg2,
                    const float* __restrict__ b_agg2,
                    float* __restrict__ out) {
  __shared__ float s_ag[NUM_CLASSES];
  int b    = blockIdx.x;
  int tid  = threadIdx.x;
  int lane = tid & 31;
  int c    = tid >> 5;                       // 0..19
  const float* h = H + ((size_t)b * NUM_CLASSES + c) * AGG_HID;
  float partial = 0.0f;
  for (int m = lane; m < AGG_HID; m += 32)
    partial += h[m] * W_agg2[m];
  #pragma unroll
  for (int off = 16; off > 0; off >>= 1)
    partial += __shfl_xor(partial, off, 32);
  if (lane == 0) {
    float ag = partial + b_agg2[0];
    s_ag[c] = ag;
    out[b * NUM_CLASSES + c] = ag;           // agreg
  }
  __syncthreads();
  if (tid == 0) {
    float mx = s_ag[0];
    int   arg = 0;
    #pragma unroll
    for (int i = 1; i < NUM_CLASSES; ++i)
      if (s_ag[i] > mx) { mx = s_ag[i]; arg = i; }   // first-max like jnp.argmax
    float e[NUM_CLASSES];
    float sum = 0.0f;
    #pragma unroll
    for (int i = 0; i < NUM_CLASSES; ++i) {
      e[i] = __expf(s_ag[i] - mx);
      sum += e[i];
    }
    float inv = 1.0f / sum;
    #pragma unroll
    for (int i = 0; i < NUM_CLASSES; ++i)
      out[BATCH * NUM_CLASSES + b * NUM_CLASSES + i] = e[i] * inv;  // prob
    out[2 * BATCH * NUM_CLASSES + b] = (float)arg;                  // class_
  }
}

extern "C" void kernel_launch(void* const* d_in, const int* in_sizes, int n_in,
                              void* d_out, int out_size, void* d_ws, size_t ws_size,
                              hipStream_t stream) {
  const int*   train      = (const int*)d_in[0];
  const float* emb_region = (const float*)d_in[1];
  const float* emb_word   = (const float*)d_in[2];
  const float* W_inter    = (const float*)d_in[3];
  const float* W_agg1     = (const float*)d_in[4];
  const float* b_agg1     = (const float*)d_in[5];
  const float* W_agg2     = (const float*)d_in[6];
  const float* b_agg2     = (const float*)d_in[7];
  float* out = (float*)d_out;

  // workspace layout (floats): P | X2P | Wp | H   (~27.9 MB total)
  float* P   = (float*)d_ws;                                   // 64*506*128
  float* X2P = P   + (size_t)BATCH * ENTIRE * EMB;             // 64*32*512
  float* Wp  = X2P + (size_t)BATCH * CLASS_PAD * ENTIRE_PAD;   // 1024*512
  float* H   = Wp  + (size_t)AGG_HID * ENTIRE_PAD;             // 1280*1024

  k_gather_max  <<<BATCH * ENTIRE, EMB,              0, stream>>>(train, emb_region, emb_word, P);
  k_pad_w       <<<AGG_HID,        128,              0, stream>>>(W_agg1, Wp);
  k_inter_wmma  <<<BATCH * 64,     32,               0, stream>>>(P, W_inter, X2P);
  k_agg1_wmma   <<<40 * 32,        32,               0, stream>>>(X2P, Wp, b_agg1, H);
  k_agg2_softmax<<<BATCH,          NUM_CLASSES * 32, 0, stream>>>(H, W_agg2, b_agg2, out);
}